// GlimpseNet_56693568307261
// MI455X (gfx1250) — compile-verified
//
#include <hip/hip_runtime.h>

// GlimpseNet forward for MI455X (gfx1250, wave32).
// extract glimpse -> WMMA GEMM1 (relu) -> fused WMMA GEMM2 + where + relu.
// V_WMMA_F32_16X16X4_F32 (fp32 in/accum) — faithful to the fp32 reference;
// ~11 MB touched / 0.19 GFLOP => latency bound; b128 staging + prefetch + LDS
// conflict-free strides are what matters here.

typedef float v2f __attribute__((ext_vector_type(2)));
typedef float v8f __attribute__((ext_vector_type(8)));

#define B_SZ    128
#define D_IN    2304
#define HG      256
#define HL      128
#define HOUT    384     // HG + HL
#define LDA_P   68      // A LDS row stride (floats): 16B aligned, ≡4 banks mod 64
#define LDB_P   80      // B LDS row stride (floats): 16B aligned, ≡16 banks mod 64

// ---------------------------------------------------------------------------
// Kernel 1: glimpse extraction + where-hidden (H2 = relu(L @ w2 + b2))
// One block per batch image, 256 threads (one per 16x16 pixel of each patch).
// ---------------------------------------------------------------------------
__global__ __launch_bounds__(256) void glimpse_extract_kernel(
    const float* __restrict__ x,   // (B,3,512,512)
    const float* __restrict__ l,   // (B,2)
    const float* __restrict__ w2,  // (2,128)
    const float* __restrict__ b2,  // (128,)
    float* __restrict__ G,         // (B,2304)
    float* __restrict__ H2)        // (B,128)
{
    const int b   = blockIdx.x;
    const int tid = threadIdx.x;            // 0..255
    const int r   = tid >> 4;
    const int col = tid & 15;

    const float l0 = l[b * 2 + 0];
    const float l1 = l[b * 2 + 1];
    // coords = (0.5*((l+1)*[512,512])).astype(int32): match fp op order, then
    // dynamic_slice clamps start to [0, padded_dim - size] == [0, 512].
    int cx = (int)(0.5f * ((l0 + 1.0f) * 512.0f));
    int cy = (int)(0.5f * ((l1 + 1.0f) * 512.0f));
    cx = min(max(cx, 0), 512);
    cy = min(max(cy, 0), 512);

    const float* __restrict__ xb = x + (size_t)b * 3 * 512 * 512;

    // 9 output channels: pc = p*3 + c, patch scale p in {0,1,2}
    #pragma unroll
    for (int pc = 0; pc < 9; ++pc) {
        const int p   = pc / 3;
        const int cch = pc % 3;
        const int k   = 1 << p;        // pool factor (1,2,4)
        const int pad = 8 << p;        // size/2 (8,16,32)
        float acc = 0.0f;
        for (int dy = 0; dy < k; ++dy) {
            const int yy = cy + r * k + dy - pad;
            for (int dx = 0; dx < k; ++dx) {
                const int xx = cx + col * k + dx - pad;
                if ((unsigned)yy < 512u && (unsigned)xx < 512u)
                    acc += xb[((size_t)cch * 512 + yy) * 512 + xx];
            }
        }
        acc *= 1.0f / (float)(k * k);
        G[((size_t)b * 9 + pc) * 256 + tid] = acc;
    }

    // where-path hidden: H2[b][j] = relu(l0*w2[0][j] + l1*w2[1][j] + b2[j])
    if (tid < HL) {
        float h = fmaf(l0, w2[tid], fmaf(l1, w2[HL + tid], b2[tid]));
        H2[b * HL + tid] = fmaxf(h, 0.0f);
    }
}

// ---------------------------------------------------------------------------
// Tiled WMMA accumulation: C_tile(16x16) += A[m0:m0+16, :K] @ B[:K, nblk+...]
// Block = 128 threads = 4 waves; block covers a 16(M) x 64(N) output slab.
// K staged through LDS in 64-wide chunks (K always a multiple of 64 here).
// f32 16x4 A layout per ISA: lanes 0-15 hold K={0,1}; lanes 16-31 hold K={2,3};
// B mirrored: VGPR j holds B[2*kh + j][n].
// ---------------------------------------------------------------------------
__device__ __forceinline__ void tile_gemm_accum(
    const float* __restrict__ A, int lda,
    const float* __restrict__ B, int ldb,
    int K, int m0, int nblk,
    float (*lA)[LDA_P], float (*lB)[LDB_P], v8f& c)
{
    const int tid  = threadIdx.x;      // 0..127
    const int lane = tid & 31;
    const int wave = tid >> 5;
    const int m    = lane & 15;
    const int kh   = lane >> 4;        // 0 or 1 -> K pair {0,1} or {2,3}
    const int nloc = wave * 16 + m;    // column within 64-wide N slab

    for (int kk = 0; kk < K; kk += 64) {
        // A chunk 16x64 floats = 256 float4; 2 per thread (b128, coalesced)
        #pragma unroll
        for (int i = 0; i < 2; ++i) {
            const int idx = i * 128 + tid;          // 0..255
            const int rr = idx >> 4;                // 0..15
            const int q4 = (idx & 15) * 4;          // 0,4,...,60
            *(float4*)&lA[rr][q4] =
                *(const float4*)&A[(size_t)(m0 + rr) * lda + kk + q4];
        }
        // B chunk 64x64 floats = 1024 float4; 8 per thread (b128, coalesced)
        #pragma unroll
        for (int i = 0; i < 8; ++i) {
            const int idx = i * 128 + tid;          // 0..1023
            const int rr = idx >> 4;                // 0..63
            const int q4 = (idx & 15) * 4;
            *(float4*)&lB[rr][q4] =
                *(const float4*)&B[(size_t)(kk + rr) * ldb + nblk + q4];
        }
        // Prefetch next B chunk (64 rows x 256B = 128 cachelines; one per thread)
        if (kk + 64 < K) {
            const int prow = tid >> 1;              // 0..63
            const int pcol = (tid & 1) * 32;        // two 128B halves per row
            __builtin_prefetch(&B[(size_t)(kk + 64 + prow) * ldb + nblk + pcol], 0, 0);
        }
        __syncthreads();

        #pragma unroll
        for (int ks = 0; ks < 64; ks += 4) {
            v2f a, bv;
            a.x  = lA[m][ks + 2 * kh + 0];
            a.y  = lA[m][ks + 2 * kh + 1];
            bv.x = lB[ks + 2 * kh + 0][nloc];
            bv.y = lB[ks + 2 * kh + 1][nloc];
            c = __builtin_amdgcn_wmma_f32_16x16x4_f32(
                    /*neg_a=*/false, a, /*neg_b=*/false, bv,
                    /*c_mod=*/(short)0, c, /*reuse_a=*/false, /*reuse_b=*/false);
        }
        __syncthreads();
    }
}

// ---------------------------------------------------------------------------
// Kernel 2: H1 = relu(G @ w1 + b1)   (128 x 2304) @ (2304 x 256)
// grid: (Nblk=4, Mtile=8), 128 threads
// ---------------------------------------------------------------------------
__global__ __launch_bounds__(128) void gemm1_kernel(
    const float* __restrict__ G, const float* __restrict__ w1,
    const float* __restrict__ b1, float* __restrict__ H1)
{
    __shared__ float lA[16][LDA_P];
    __shared__ float lB[64][LDB_P];
    const int m0   = blockIdx.y * 16;
    const int nblk = blockIdx.x * 64;

    v8f c = {0.f, 0.f, 0.f, 0.f, 0.f, 0.f, 0.f, 0.f};
    tile_gemm_accum(G, D_IN, w1, HG, D_IN, m0, nblk, lA, lB, c);

    const int lane = threadIdx.x & 31;
    const int wave = threadIdx.x >> 5;
    const int kh   = lane >> 4;
    const int n    = nblk + wave * 16 + (lane & 15);
    const float bias = b1[n];
    // 16x16 f32 C/D layout: VGPR r -> M = r (lanes 0-15) / r+8 (lanes 16-31)
    #pragma unroll
    for (int r = 0; r < 8; ++r) {
        const int mm = m0 + r + 8 * kh;
        H1[(size_t)mm * HG + n] = fmaxf(c[r] + bias, 0.0f);
    }
}

// ---------------------------------------------------------------------------
// Kernel 3: out = relu(H1 @ w3 + b3 + H2 @ w4 + b4)   -> (128 x 384)
// grid: (Nblk=6, Mtile=8), 128 threads
// ---------------------------------------------------------------------------
__global__ __launch_bounds__(128) void gemm2_kernel(
    const float* __restrict__ H1, const float* __restrict__ w3,
    const float* __restrict__ b3,
    const float* __restrict__ H2, const float* __restrict__ w4,
    const float* __restrict__ b4, float* __restrict__ out)
{
    __shared__ float lA[16][LDA_P];
    __shared__ float lB[64][LDB_P];
    const int m0   = blockIdx.y * 16;
    const int nblk = blockIdx.x * 64;

    v8f c = {0.f, 0.f, 0.f, 0.f, 0.f, 0.f, 0.f, 0.f};
    tile_gemm_accum(H1, HG, w3, HOUT, HG, m0, nblk, lA, lB, c);  // K = 256
    tile_gemm_accum(H2, HL, w4, HOUT, HL, m0, nblk, lA, lB, c);  // K = 128

    const int lane = threadIdx.x & 31;
    const int wave = threadIdx.x >> 5;
    const int kh   = lane >> 4;
    const int n    = nblk + wave * 16 + (lane & 15);
    const float bias = b3[n] + b4[n];
    #pragma unroll
    for (int r = 0; r < 8; ++r) {
        const int mm = m0 + r + 8 * kh;
        out[(size_t)mm * HOUT + n] = fmaxf(c[r] + bias, 0.0f);
    }
}

// ---------------------------------------------------------------------------
extern "C" void kernel_launch(void* const* d_in, const int* in_sizes, int n_in,
                              void* d_out, int out_size, void* d_ws, size_t ws_size,
                              hipStream_t stream) {
    const float* x  = (const float*)d_in[0];   // (128,3,512,512)
    const float* l  = (const float*)d_in[1];   // (128,2)
    const float* w1 = (const float*)d_in[2];   // (2304,256)
    const float* b1 = (const float*)d_in[3];   // (256,)
    const float* w2 = (const float*)d_in[4];   // (2,128)
    const float* b2 = (const float*)d_in[5];   // (128,)
    const float* w3 = (const float*)d_in[6];   // (256,384)
    const float* b3 = (const float*)d_in[7];   // (384,)
    const float* w4 = (const float*)d_in[8];   // (128,384)
    const float* b4 = (const float*)d_in[9];   // (384,)
    float* out = (float*)d_out;                // (128,384)

    float* G  = (float*)d_ws;                  // (128,2304)
    float* H1 = G + (size_t)B_SZ * D_IN;       // (128,256)
    float* H2 = H1 + (size_t)B_SZ * HG;        // (128,128)

    glimpse_extract_kernel<<<dim3(B_SZ), dim3(256), 0, stream>>>(x, l, w2, b2, G, H2);
    gemm1_kernel<<<dim3(HG / 64, B_SZ / 16), dim3(128), 0, stream>>>(G, w1, b1, H1);
    gemm2_kernel<<<dim3(HOUT / 64, B_SZ / 16), dim3(128), 0, stream>>>(H1, w3, b3, H2, w4, b4, out);
}